// TreeLRU_670014899093
// MI455X (gfx1250) — compile-verified
//
#include <hip/hip_runtime.h>
#include <hip/hip_bf16.h>

// CDNA5 (gfx1250) tree-LRU using V_WMMA_F32_16X16X4_F32 for all fp32 GEMMs.

typedef __attribute__((ext_vector_type(2))) float v2f;
typedef __attribute__((ext_vector_type(4))) float v4f;
typedef __attribute__((ext_vector_type(8))) float v8f;

#define NF 128          // IN_F == OUT_F == ST_F
#define NNODES 8191
#define BSZ 32
#define LSTR 132        // LDS row stride (floats), padded: 132 % 64 == 4 -> conflict-free frags

__device__ __forceinline__ v8f wmma_f32(v2f a, v2f b, v8f c) {
    return __builtin_amdgcn_wmma_f32_16x16x4_f32(
        /*neg_a=*/false, a, /*neg_b=*/false, b,
        /*c_mod=*/(short)0, c, /*reuse_a=*/false, /*reuse_b=*/false);
}

// ---------------------------------------------------------------------------
// proj = x @ Wp^T + bp        (262112 x 128) = (262112 x 128)(128 x 128)^T
// One wave per 16-row tile; 8 N-tiles of 16; K-loop of 32 x WMMA(16x16x4).
// ---------------------------------------------------------------------------
__global__ __launch_bounds__(32)
void proj_gemm_kernel(const float* __restrict__ x,
                      const float* __restrict__ Wp,
                      const float* __restrict__ bp,
                      float* __restrict__ proj)
{
    const int lane  = threadIdx.x;
    const int rsel  = lane & 15;      // row (A) / col (B) selector
    const int halfk = lane >> 4;      // K-pair selector
    const int rowBase = blockIdx.x * 16;

    const float* xrow = x + (size_t)(rowBase + rsel) * NF;

#pragma unroll
    for (int nt = 0; nt < 8; ++nt) {
        v8f acc = {};
        const float* wrow = Wp + (nt * 16 + rsel) * NF;
#pragma unroll 4
        for (int k0 = 0; k0 < NF; k0 += 4) {
            const int kb = k0 + 2 * halfk;
            v2f a = *(const v2f*)(xrow + kb);
            v2f b = *(const v2f*)(wrow + kb);
            acc = wmma_f32(a, b, acc);
        }
        const float bias = bp[nt * 16 + rsel];
#pragma unroll
        for (int v = 0; v < 8; ++v) {
            const int m = rowBase + v + 8 * halfk;          // C/D layout: row = v + 8*(lane>=16)
            proj[(size_t)m * NF + nt * 16 + rsel] = acc[v] + bias;
        }
    }
}

// ---------------------------------------------------------------------------
// One tree level: for 16 (batch,node) rows per wave:
//   it   = proj_row @ Bm^T                (complex: 2 WMMA chains)
//   h    = lam*(st[l]+st[r]) + gamma*it   (elementwise, lam/gamma from logs)
//   y    = h_re @ CmRe^T - h_im @ CmIm^T  (2 WMMA chains, via LDS restage)
// ---------------------------------------------------------------------------
__global__ __launch_bounds__(32)
void lru_level_kernel(const float* __restrict__ proj,
                      const float* __restrict__ Bm,     // complex interleaved (ST_F, IN_F)
                      const float* __restrict__ Cm,     // complex interleaved (OUT_F, ST_F)
                      const float* __restrict__ nu_log,
                      const float* __restrict__ th_log,
                      const float* __restrict__ ga_log,
                      const int* __restrict__ sb, const int* __restrict__ sn,
                      const int* __restrict__ sl, const int* __restrict__ sr,
                      float* __restrict__ stRe, float* __restrict__ stIm,
                      float* __restrict__ out,
                      int off)
{
    __shared__ int   s_row[16];                // (b*NNODES + n) * NF : row offset in proj/state/out
    __shared__ int   s_bb[16];                 // b*NNODES*NF : state-plane base for child gathers
    __shared__ int   s_l[16], s_r[16];
    __shared__ float s_cre[16 * LSTR];
    __shared__ float s_cim[16 * LSTR];
    __shared__ float s_hre[16 * LSTR];
    __shared__ float s_him[16 * LSTR];

    const int lane  = threadIdx.x;
    const int rsel  = lane & 15;
    const int halfk = lane >> 4;

    // --- stage schedule indices for this 16-row tile ---
    if (lane < 16) {
        const int r = off + blockIdx.x * 16 + lane;
        const int b = sb[r];
        const int n = sn[r];
        s_row[lane] = (b * NNODES + n) * NF;
        s_bb[lane]  = b * NNODES * NF;
        s_l[lane]   = sl[r];
        s_r[lane]   = sr[r];
    }
    __syncthreads();

    // --- cooperative gather of child states (re+im) into LDS ---
    for (int m = 0; m < 16; ++m) {
        const int l  = s_l[m];
        const int rr = s_r[m];
        const int bb = s_bb[m];
#pragma unroll
        for (int c = lane; c < NF; c += 32) {
            float cre = 0.f, cim = 0.f;
            if (l >= 0)  { cre += stRe[bb + l  * NF + c]; cim += stIm[bb + l  * NF + c]; }
            if (rr >= 0) { cre += stRe[bb + rr * NF + c]; cim += stIm[bb + rr * NF + c]; }
            s_cre[m * LSTR + c] = cre;
            s_cim[m * LSTR + c] = cim;
        }
    }
    __syncthreads();

    const float* projRow = proj + s_row[rsel];

    // --- Bm matmul + fused recurrence; h -> LDS and global states ---
#pragma unroll 1
    for (int nt = 0; nt < 8; ++nt) {
        v8f aRe = {}, aIm = {};
        const float* brow = Bm + (size_t)(nt * 16 + rsel) * NF * 2;   // interleaved complex row
#pragma unroll 4
        for (int k0 = 0; k0 < NF; k0 += 4) {
            const int kb = k0 + 2 * halfk;
            v2f afrag = *(const v2f*)(projRow + kb);
            v4f bq    = *(const v4f*)(brow + kb * 2);                 // re0,im0,re1,im1
            v2f bre = { bq.x, bq.z };
            v2f bim = { bq.y, bq.w };
            aRe = wmma_f32(afrag, bre, aRe);
            aIm = wmma_f32(afrag, bim, aIm);
        }
        // per-lane feature column
        const int f = nt * 16 + rsel;
        const float lam_mod = __expf(-__expf(nu_log[f]));
        const float theta   = __expf(th_log[f]);
        const float lamRe   = lam_mod * __cosf(theta);
        const float lamIm   = lam_mod * __sinf(theta);
        const float ga      = __expf(ga_log[f]);
#pragma unroll
        for (int v = 0; v < 8; ++v) {
            const int m = v + 8 * halfk;
            const float cre = s_cre[m * LSTR + f];
            const float cim = s_cim[m * LSTR + f];
            const float hre = lamRe * cre - lamIm * cim + ga * aRe[v];
            const float him = lamRe * cim + lamIm * cre + ga * aIm[v];
            s_hre[m * LSTR + f] = hre;
            s_him[m * LSTR + f] = him;
            stRe[s_row[m] + f] = hre;       // parents at next level read these
            stIm[s_row[m] + f] = him;
        }
    }
    __syncthreads();

    // --- Cm matmul: y = h_re @ CmRe^T - h_im @ CmIm^T ---
#pragma unroll 1
    for (int nt = 0; nt < 8; ++nt) {
        v8f acc = {}, acc2 = {};
        const float* crow = Cm + (size_t)(nt * 16 + rsel) * NF * 2;
#pragma unroll 4
        for (int k0 = 0; k0 < NF; k0 += 4) {
            const int kb = k0 + 2 * halfk;
            v2f hre = *(const v2f*)(&s_hre[rsel * LSTR + kb]);
            v2f him = *(const v2f*)(&s_him[rsel * LSTR + kb]);
            v4f cq  = *(const v4f*)(crow + kb * 2);
            v2f cre = { cq.x, cq.z };
            v2f cim = { cq.y, cq.w };
            acc  = wmma_f32(hre, cre, acc);
            acc2 = wmma_f32(him, cim, acc2);
        }
#pragma unroll
        for (int v = 0; v < 8; ++v) {
            const int m = v + 8 * halfk;
            out[s_row[m] + nt * 16 + rsel] = acc[v] - acc2[v];
        }
    }
}

extern "C" void kernel_launch(void* const* d_in, const int* in_sizes, int n_in,
                              void* d_out, int out_size, void* d_ws, size_t ws_size,
                              hipStream_t stream) {
    const float* x   = (const float*)d_in[0];
    const float* Wp  = (const float*)d_in[1];
    const float* bp  = (const float*)d_in[2];
    const float* Bm  = (const float*)d_in[3];   // complex64 -> interleaved float pairs
    const float* Cm  = (const float*)d_in[4];
    const float* nu  = (const float*)d_in[5];
    const float* th  = (const float*)d_in[6];
    const float* ga  = (const float*)d_in[7];
    const int*   sb  = (const int*)d_in[8];
    const int*   sn  = (const int*)d_in[9];
    const int*   sl  = (const int*)d_in[10];
    const int*   sr  = (const int*)d_in[11];
    float* out = (float*)d_out;

    const size_t plane = (size_t)BSZ * NNODES * NF;   // 33,550,336 floats
    float* proj  = (float*)d_ws;
    float* stRe  = proj + plane;
    float* stIm  = stRe + plane;

    const int totalRows = BSZ * NNODES;               // 262,112 (divisible by 16)
    proj_gemm_kernel<<<totalRows / 16, 32, 0, stream>>>(x, Wp, bp, proj);

    int off = 0;
    for (int h = 0; h < 13; ++h) {                    // leaves first
        const int cnt = BSZ << (12 - h);              // 131072 ... 32
        lru_level_kernel<<<cnt / 16, 32, 0, stream>>>(proj, Bm, Cm, nu, th, ga,
                                                      sb, sn, sl, sr,
                                                      stRe, stIm, out, off);
        off += cnt;
    }
}